// DriftingPolicy_30296699306003
// MI455X (gfx1250) — compile-verified
//
#include <hip/hip_runtime.h>

typedef __attribute__((ext_vector_type(2))) float v2f;
typedef __attribute__((ext_vector_type(8))) float v8f;

#define NB     4096     // number of points
#define DIM    128      // flattened feature dim (16*8)
#define MBLK   64       // rows of x per workgroup (4 waves x 16)
#define JBLK   16       // y rows per j-block
#define NWAVE  4
#define YROW_S 132      // padded stride (floats) for row-major y block
#define YT_S   18       // padded stride (floats) for transposed y block
#define W_S    18       // padded stride (floats) for per-wave w tile

// ds_swizzle with xor-mask M within 32 lanes (and_mask=0x1f, or_mask=0)
template <int IMM>
__device__ __forceinline__ float swz(float v) {
  return __int_as_float(__builtin_amdgcn_ds_swizzle(__float_as_int(v), IMM));
}
#define SWZ_XOR(m) ((((m) & 31) << 10) | 0x1f)

__global__ __launch_bounds__(128) void drift_kernel(
    const float* __restrict__ x, const float* __restrict__ yP,
    const float* __restrict__ yN, float* __restrict__ dstBase,
    int dstFieldStride, int fieldParam, int accumulate) {
  __shared__ float y_row[JBLK * YROW_S];
  __shared__ float y_t[DIM * YT_S];
  __shared__ float y_sq[JBLK];
  __shared__ float w_lds[NWAVE * 16 * W_S];

  const int tid  = threadIdx.x;
  const int wvid = tid >> 5;
  const int lane = tid & 31;
  const int h    = lane >> 4;   // half of wave (selects K/M sub-block)
  const int l16  = lane & 15;

  const int field = (gridDim.y == 2) ? (int)blockIdx.y : fieldParam;
  const float* __restrict__ y = field ? yN : yP;
  const float coef = field ? 0.5f : -1.0f;  // NEG_DRIFT*(+1) / DRIFT*(-1)
  const int rbase = blockIdx.x * MBLK + wvid * 16;
  float* wmine = &w_lds[wvid * 16 * W_S];

  // ---- load x strip as A-fragments (16x4 f32 A layout: lane<16 -> K=2h+{0,1})
  v2f xfrag[32];
  float xpart = 0.f;
#pragma unroll
  for (int c = 0; c < 32; ++c) {
    const float* p = x + (size_t)(rbase + l16) * DIM + 4 * c + 2 * h;
    v2f t;
    t.x = p[0];
    t.y = p[1];
    xfrag[c] = t;
    xpart += t.x * t.x + t.y * t.y;
  }
  // full ||x_row||^2: fold the two wave halves
  float xsq_row = xpart + swz<SWZ_XOR(16)>(xpart);
  // per-lane x_sq for the 8 D-layout rows (r + 8*h)
  float xs[8];
#pragma unroll
  for (int r = 0; r < 8; ++r)
    xs[r] = __int_as_float(
        __builtin_amdgcn_ds_bpermute((r + 8 * h) * 4, __float_as_int(xsq_row)));

  v8f acc[8];
#pragma unroll
  for (int t = 0; t < 8; ++t) acc[t] = (v8f){0, 0, 0, 0, 0, 0, 0, 0};
  float srow[8];
#pragma unroll
  for (int r = 0; r < 8; ++r) srow[r] = 0.f;

#pragma clang loop unroll(disable)
  for (int jb = 0; jb < NB / JBLK; ++jb) {
    __syncthreads();  // previous iteration's LDS readers done

    // ---- stage y block: row-major + transposed + row norms
#pragma unroll
    for (int k = 0; k < 4; ++k) {
      int f = tid + 128 * k;       // float4 unit id, 512 total
      int row = f >> 5;            // 0..15 ; one full row per (wave,k)
      int c4 = f & 31;
      const float4 v =
          *(const float4*)(y + (size_t)(jb * JBLK + row) * DIM + c4 * 4);
      *(float4*)&y_row[row * YROW_S + c4 * 4] = v;
      y_t[(c4 * 4 + 0) * YT_S + row] = v.x;
      y_t[(c4 * 4 + 1) * YT_S + row] = v.y;
      y_t[(c4 * 4 + 2) * YT_S + row] = v.z;
      y_t[(c4 * 4 + 3) * YT_S + row] = v.w;
      float p = v.x * v.x + v.y * v.y + v.z * v.z + v.w * v.w;
      p += swz<SWZ_XOR(1)>(p);
      p += swz<SWZ_XOR(2)>(p);
      p += swz<SWZ_XOR(4)>(p);
      p += swz<SWZ_XOR(8)>(p);
      p += swz<SWZ_XOR(16)>(p);
      if (lane == 0) y_sq[row] = p;
    }
    __syncthreads();

    // ---- GEMM1: S = x_strip @ y_block^T   (K = 128, f32 WMMA 16x16x4)
    // Two independent accumulation chains for ILP, merged at the end.
    v8f S0 = (v8f){0, 0, 0, 0, 0, 0, 0, 0};
    v8f S1 = (v8f){0, 0, 0, 0, 0, 0, 0, 0};
#pragma unroll
    for (int c = 0; c < 32; c += 2) {
      v2f bf0 = *(const v2f*)&y_row[l16 * YROW_S + 4 * c + 2 * h];
      v2f bf1 = *(const v2f*)&y_row[l16 * YROW_S + 4 * (c + 1) + 2 * h];
      S0 = __builtin_amdgcn_wmma_f32_16x16x4_f32(false, xfrag[c], false, bf0,
                                                 (short)0, S0, false, false);
      S1 = __builtin_amdgcn_wmma_f32_16x16x4_f32(false, xfrag[c + 1], false, bf1,
                                                 (short)0, S1, false, false);
    }

    // ---- w = exp(-max(d2,0)/2), mask diagonal, accumulate rowsums
    const float ysq_l = y_sq[l16];
    const int colg = jb * JBLK + l16;
#pragma unroll
    for (int r = 0; r < 8; ++r) {
      const float Sr = S0[r] + S1[r];
      float d2 = xs[r] + ysq_l - 2.f * Sr;
      d2 = fmaxf(d2, 0.f);
      float wval = __builtin_amdgcn_exp2f(d2 * -0.72134752044448170f);
      const int rowg = rbase + r + 8 * h;
      if (rowg == colg) wval = 0.f;
      srow[r] += wval;
      wmine[(r + 8 * h) * W_S + l16] = wval;  // D-layout -> LDS row-major
    }
    __builtin_amdgcn_wave_barrier();  // keep w stores before re-reads

    // ---- GEMM2: acc += w @ y_block    (K = 16 over j, N = 128 features)
    // 8 independent acc[t] chains (4 deep each) -> plenty of ILP.
    v2f wA[4];
#pragma unroll
    for (int c2 = 0; c2 < 4; ++c2)
      wA[c2] = *(const v2f*)&wmine[l16 * W_S + 4 * c2 + 2 * h];
#pragma unroll
    for (int t = 0; t < 8; ++t) {
      const int nb = (16 * t + l16) * YT_S + 2 * h;
      v2f b0 = *(const v2f*)&y_t[nb + 0];
      v2f b1 = *(const v2f*)&y_t[nb + 4];
      v2f b2 = *(const v2f*)&y_t[nb + 8];
      v2f b3 = *(const v2f*)&y_t[nb + 12];
      acc[t] = __builtin_amdgcn_wmma_f32_16x16x4_f32(false, wA[0], false, b0,
                                                     (short)0, acc[t], false, false);
      acc[t] = __builtin_amdgcn_wmma_f32_16x16x4_f32(false, wA[1], false, b1,
                                                     (short)0, acc[t], false, false);
      acc[t] = __builtin_amdgcn_wmma_f32_16x16x4_f32(false, wA[2], false, b2,
                                                     (short)0, acc[t], false, false);
      acc[t] = __builtin_amdgcn_wmma_f32_16x16x4_f32(false, wA[3], false, b3,
                                                     (short)0, acc[t], false, false);
    }
  }

  // ---- reduce rowsums across the 16 lanes of each half, form ratio/inv
  float ratio[8], inv[8];
#pragma unroll
  for (int r = 0; r < 8; ++r) {
    float s = srow[r];
    s += swz<SWZ_XOR(1)>(s);
    s += swz<SWZ_XOR(2)>(s);
    s += swz<SWZ_XOR(4)>(s);
    s += swz<SWZ_XOR(8)>(s);
    const float den = fmaxf(s, 1e-8f);
    ratio[r] = s / den;
    inv[r] = 1.f / den;
  }

  // ---- write field contribution: coef * (x*ratio - inv*acc)
  float* dst = dstBase + (size_t)field * (size_t)dstFieldStride;
#pragma unroll
  for (int t = 0; t < 8; ++t) {
#pragma unroll
    for (int r = 0; r < 8; ++r) {
      const int rowg = rbase + r + 8 * h;
      const int col = 16 * t + l16;
      const size_t idx = (size_t)rowg * DIM + col;
      const float xval = x[idx];
      const float res = coef * (xval * ratio[r] - inv[r] * acc[t][r]);
      if (accumulate)
        dst[idx] += res;
      else
        dst[idx] = res;
    }
  }
}

__global__ void combine_kernel(const float* __restrict__ a,
                               const float* __restrict__ b,
                               float* __restrict__ o, int n4) {
  int i = blockIdx.x * blockDim.x + threadIdx.x;
  if (i < n4) {
    float4 va = ((const float4*)a)[i];
    float4 vb = ((const float4*)b)[i];
    float4 r = {va.x + vb.x, va.y + vb.y, va.z + vb.z, va.w + vb.w};
    ((float4*)o)[i] = r;
  }
}

extern "C" void kernel_launch(void* const* d_in, const int* in_sizes, int n_in,
                              void* d_out, int out_size, void* d_ws,
                              size_t ws_size, hipStream_t stream) {
  const float* x  = (const float*)d_in[0];
  const float* yP = (const float*)d_in[1];
  const float* yN = (const float*)d_in[2];
  float* out = (float*)d_out;
  const size_t N = (size_t)NB * DIM;  // 524288 == out_size

  if (ws_size >= 2 * N * sizeof(float)) {
    // both fields concurrently -> workspace partials, then combine
    float* ws = (float*)d_ws;
    dim3 grid(NB / MBLK, 2);
    drift_kernel<<<grid, 128, 0, stream>>>(x, yP, yN, ws, (int)N, 0, 0);
    const int n4 = (int)(N / 4);
    combine_kernel<<<(n4 + 255) / 256, 256, 0, stream>>>(ws, ws + N, out, n4);
  } else {
    // fallback: serialize fields directly into out (no workspace needed)
    dim3 grid(NB / MBLK, 1);
    drift_kernel<<<grid, 128, 0, stream>>>(x, yP, yN, out, 0, 0, 0);
    drift_kernel<<<grid, 128, 0, stream>>>(x, yP, yN, out, 0, 1, 1);
  }
}